// LSR_QHNetBackBoneSO2_symmetry_84748294684786
// MI455X (gfx1250) — compile-verified
//
#include <hip/hip_runtime.h>
#include <hip/hip_bf16.h>

// ---------------------------------------------------------------------------
// Problem constants (match reference)
// ---------------------------------------------------------------------------
#define BMOL 256
#define MATM 48
#define NTOT (BMOL * MATM)            // 12288 atoms
#define EPM  (MATM * (MATM - 1))      // 2256 edges per molecule
#define ETOT (BMOL * EPM)             // 577536 edges
#define KRBF 32
#define NSH  25
#define FDIM (KRBF + NSH)             // 57
#define HS   128
#define NTYPES 20

typedef __attribute__((ext_vector_type(2))) float v2f;
typedef __attribute__((ext_vector_type(8))) float v8f;

// ---------------------------------------------------------------------------
// Kernel 1: per-edge features (RBF + SH) + edge_index + transpose_index.
// LDS-staged so feat rows go out as coalesced 128-bit stores.
// ---------------------------------------------------------------------------
__global__ __launch_bounds__(256) void edge_feat_kernel(
    const float* __restrict__ pos,
    const float* __restrict__ alpha,
    float* __restrict__ feat,
    int* __restrict__ eidx,          // [2, E] : dst row then src row
    int* __restrict__ tidx) {        // [E]
  __shared__ float sfeat[256 * FDIM];   // 58368 B (within 320 KB WGP LDS)

  const int tid = threadIdx.x;
  const int e   = blockIdx.x * 256 + tid;      // grid sized exactly: E/256 blocks

  // Decode edge (src-major complete digraph per molecule)
  const int g = e / EPM;
  const int t = e - g * EPM;
  const int s = t / (MATM - 1);
  const int k = t - s * (MATM - 1);
  const int d = k + (k >= s ? 1 : 0);
  const int src = g * MATM + s;
  const int dst = g * MATM + d;
  const int tr  = g * EPM + d * (MATM - 1) + s - (d < s ? 1 : 0);

  eidx[e]        = dst;
  eidx[ETOT + e] = src;
  tidx[e]        = tr;

  // Edge vector & unit direction
  const float dx = pos[dst * 3 + 0] - pos[src * 3 + 0];
  const float dy = pos[dst * 3 + 1] - pos[src * 3 + 1];
  const float dz = pos[dst * 3 + 2] - pos[src * 3 + 2];
  float r = sqrtf(dx * dx + dy * dy + dz * dz);
  r = fmaxf(r, 1e-6f);
  const float inv = 1.0f / r;
  const float x = dx * inv, y = dy * inv, z = dz * inv;

  float* row = &sfeat[tid * FDIM];

  // ---- Exponential Bernstein RBF (log-domain, matches reference) ----
  const float al  = 0.5f * alpha[0];
  const float xx  = -al * r;                    // xx < 0
  const float lem = logf(-expm1f(xx));          // log(1 - e^xx)
  const float rc  = r * (1.0f / 15.0f);
  const float den = fmaxf((1.0f - rc) * (1.0f + rc), 1e-9f);
  const float fcut = (rc < 1.0f) ? expf(-rc * rc / den) : 0.0f;

  float lb = 0.0f;                              // running log-binomial C(31, v)
  #pragma unroll
  for (int v = 0; v < KRBF; ++v) {
    row[v] = fcut * expf(lb + (float)(KRBF - 1 - v) * xx + (float)v * lem);
    if (v < KRBF - 1)
      lb += logf((float)(KRBF - 1 - v) / (float)(v + 1));  // constant-folds under unroll
  }

  // ---- Real spherical harmonics, component norm, up to l=4 ----
  const float C1 = x,              S1 = y;
  const float C2 = C1 * x - S1 * y, S2 = S1 * x + C1 * y;
  const float C3 = C2 * x - S2 * y, S3 = S2 * x + C2 * y;
  const float C4 = C3 * x - S3 * y, S4 = S3 * x + C3 * y;

  const float Q10 = z;
  const float Q20 = 1.5f * z * z - 0.5f;
  const float Q30 = (5.0f * z * Q20 - 2.0f * Q10) * (1.0f / 3.0f);
  const float Q40 = (7.0f * z * Q30 - 3.0f * Q20) * 0.25f;
  const float Q21 = 3.0f * z;
  const float Q31 = (5.0f * z * Q21 - 3.0f) * 0.5f;
  const float Q41 = (7.0f * z * Q31 - 4.0f * Q21) * (1.0f / 3.0f);
  const float Q32 = 15.0f * z;
  const float Q42 = (7.0f * z * Q32 - 15.0f) * 0.5f;

  row[32] = 1.0f;
  row[33] = 1.7320508075688772f * S1;
  row[34] = 1.7320508075688772f * Q10;
  row[35] = 1.7320508075688772f * C1;

  row[36] = 1.9364916731037085f * S2;           // c22*Q22
  row[37] = 3.8729833462074170f * z * S1;       // c21*Q21
  row[38] = 2.2360679774997896f * Q20;
  row[39] = 3.8729833462074170f * z * C1;
  row[40] = 1.9364916731037085f * C2;

  row[41] = 2.0916500663351889f * S3;           // c33*Q33
  row[42] = 5.1234753829797990f * z * S2;       // c32*Q32
  row[43] = 1.0801234497346435f * Q31 * S1;
  row[44] = 2.6457513110645906f * Q30;
  row[45] = 1.0801234497346435f * Q31 * C1;
  row[46] = 5.1234753829797990f * z * C2;
  row[47] = 2.0916500663351889f * C3;

  row[48] = 2.2185299186623563f * S4;           // c44*Q44
  row[49] = 6.2749501990055660f * z * S3;       // c43*Q43
  row[50] = 0.2236067977499790f * Q42 * S2;
  row[51] = 0.9486832980505138f * Q41 * S1;
  row[52] = 3.0f * Q40;
  row[53] = 0.9486832980505138f * Q41 * C1;
  row[54] = 0.2236067977499790f * Q42 * C2;
  row[55] = 6.2749501990055660f * z * C3;
  row[56] = 2.2185299186623563f * C4;

  __syncthreads();

  // Coalesced b128 stores: block's LDS image == contiguous global slab
  const float4* s4 = (const float4*)sfeat;
  float4* g4 = (float4*)(feat + (size_t)blockIdx.x * (256 * FDIM));
  #pragma unroll
  for (int i = tid; i < (256 * FDIM) / 4; i += 256)
    g4[i] = s4[i];
}

// ---------------------------------------------------------------------------
// Kernel 2: node_emb = OneHot(atomic_numbers)[N,20] @ table[20,128]
// via V_WMMA_F32_16X16X4_F32 (exact in f32: one-hot A selects table rows).
// One wave per 16x16 output tile; 5 chained WMMAs cover K = 20 types.
// Block = 256 threads (8 waves) = one 16-row x 128-col slab.
// ---------------------------------------------------------------------------
__global__ __launch_bounds__(256) void embed_wmma_kernel(
    const int* __restrict__ an,
    const float* __restrict__ table,   // [20, 128]
    float* __restrict__ out) {         // [N, 128]
  const int lane    = threadIdx.x & 31;
  const int wave    = threadIdx.x >> 5;
  const int rowBase = blockIdx.x * 16;     // gridDim.x = N/16 = 768
  const int colBase = wave * 16;           // 8 waves -> 128 cols
  const int mrow    = lane & 15;
  const int hi      = (lane >= 16) ? 1 : 0;

  // This lane's A-matrix row type (A 16x4 f32 layout: lanes 0-15 K={0,1},
  // lanes 16-31 K={2,3}; row = lane%16)
  const int myType = an[rowBase + mrow];

  v8f acc = {};
  #pragma unroll
  for (int kb = 0; kb < NTYPES; kb += 4) {
    const int k0 = kb + (hi ? 2 : 0);
    const int k1 = k0 + 1;
    v2f a;
    a.x = (myType == k0) ? 1.0f : 0.0f;
    a.y = (myType == k1) ? 1.0f : 0.0f;
    v2f b;   // B 4x16 f32: same lane split; col = lane%16
    b.x = table[k0 * HS + colBase + mrow];
    b.y = table[k1 * HS + colBase + mrow];
    acc = __builtin_amdgcn_wmma_f32_16x16x4_f32(
        /*neg_a=*/false, a, /*neg_b=*/false, b,
        /*c_mod=*/(short)0, acc, /*reuse_a=*/false, /*reuse_b=*/false);
  }

  // C/D 16x16 f32 layout: VGPR v -> row v (lanes 0-15) / v+8 (lanes 16-31)
  #pragma unroll
  for (int v = 0; v < 8; ++v) {
    const int row = rowBase + v + (hi ? 8 : 0);
    out[(size_t)row * HS + colBase + mrow] = acc[v];
  }
}

// ---------------------------------------------------------------------------
// Launch: d_out = [feat E*57 | node_emb N*128 | edge_index 2*E | transpose E]
// ---------------------------------------------------------------------------
extern "C" void kernel_launch(void* const* d_in, const int* in_sizes, int n_in,
                              void* d_out, int out_size, void* d_ws, size_t ws_size,
                              hipStream_t stream) {
  const float* pos   = (const float*)d_in[0];
  const int*   an    = (const int*)d_in[1];
  const float* table = (const float*)d_in[2];
  const float* alpha = (const float*)d_in[3];

  float* out  = (float*)d_out;
  float* feat = out;
  float* node = out + (size_t)ETOT * FDIM;
  int*   eidx = (int*)(node + (size_t)NTOT * HS);
  int*   tidx = eidx + 2 * (size_t)ETOT;

  edge_feat_kernel<<<ETOT / 256, 256, 0, stream>>>(pos, alpha, feat, eidx, tidx);
  embed_wmma_kernel<<<NTOT / 16, 256, 0, stream>>>(an, table, node);
}